// Generator_causal_6777458393268
// MI455X (gfx1250) — compile-verified
//
#include <hip/hip_runtime.h>
#include <math.h>

// ---------------------------------------------------------------------------
// Causal generator scan, MI455X (gfx1250), full fp32 via V_WMMA_F32_16X16X4_F32
// ---------------------------------------------------------------------------

typedef __attribute__((ext_vector_type(2))) float v2f;
typedef __attribute__((ext_vector_type(8))) float v8f;

#define BATCH 65536
#define XD 32
#define HD 128
#define THREADS 256
#define WAVES 8
#define ROWS_PER_BLOCK 128   // 8 waves * 16 rows

__global__ __launch_bounds__(THREADS) void causal_gen_kernel(
    const float* __restrict__ x,  const float* __restrict__ z,
    const float* __restrict__ M,  const float* __restrict__ Wi,
    const float* __restrict__ bi, const float* __restrict__ wf,
    const float* __restrict__ bf, const float* __restrict__ W1,
    const float* __restrict__ b1, const float* __restrict__ W2,
    const float* __restrict__ b2, float* __restrict__ out)
{
    // B-layout packed weights: Bp[(k>>1)*128 + n] = (W[n][k], W[n][k+1])
    __shared__ v2f  W1p[64 * 128];          // 64 KB
    __shared__ v2f  W2p[64 * 128];          // 64 KB
    __shared__ v2f  Wip[16 * 128];          // 16 KB (per-step, x-part of Wi_i)
    __shared__ float Mt[32 * 32];           // Mt[i*32+k] = M[k][i]
    __shared__ float Wiz[HD], biL[HD], b1L[HD], b2L[HD], wfL[HD];
    __shared__ __align__(16) float hscr[WAVES][16 * HD];   // 64 KB transpose scratch
    __shared__ float valL[WAVES][16];

    const int tid  = threadIdx.x;
    const int wave = tid >> 5;
    const int lane = tid & 31;
    const int ln   = lane & 15;   // A-row / B-col within tile
    const int hh   = lane >> 4;   // which K-half of a K=4 chunk
    const int rowBase = blockIdx.x * ROWS_PER_BLOCK + wave * 16;

    // ---- one-time: pack W1/W2 into B-layout, stage Mt and biases ----
    for (int idx = tid; idx < 128 * 64; idx += THREADS) {
        int n = idx >> 6, k2 = idx & 63;
        v2f w1v, w2v;
        w1v.x = W1[n * HD + 2 * k2]; w1v.y = W1[n * HD + 2 * k2 + 1];
        w2v.x = W2[n * HD + 2 * k2]; w2v.y = W2[n * HD + 2 * k2 + 1];
        W1p[k2 * 128 + n] = w1v;
        W2p[k2 * 128 + n] = w2v;
    }
    for (int idx = tid; idx < 32 * 32; idx += THREADS) {
        int i = idx >> 5, k = idx & 31;
        Mt[idx] = M[k * XD + i];
    }
    if (tid < HD) { b1L[tid] = b1[tid]; b2L[tid] = b2[tid]; }
    __syncthreads();

    // ---- out-state in registers, WMMA A-layout:
    //      out_a[c].x = out[row=ln][4c+2hh], .y = [4c+2hh+1]
    v2f out_a[8];
#pragma unroll
    for (int c = 0; c < 8; ++c)
        out_a[c] = *(const v2f*)(x + (rowBase + ln) * XD + 4 * c + 2 * hh);

    for (int i = 0; i < XD; ++i) {
        // ---- stage step-i weights into LDS (packed B-layout) ----
        const float* wib = Wi + (size_t)i * HD * (XD + 1);
        for (int idx = tid; idx < 128 * 16; idx += THREADS) {
            int n = idx >> 4, k2 = idx & 15;
            v2f w;
            w.x = wib[n * (XD + 1) + 2 * k2];
            w.y = wib[n * (XD + 1) + 2 * k2 + 1];
            Wip[k2 * 128 + n] = w;
        }
        if (tid < HD) {
            Wiz[tid] = wib[tid * (XD + 1) + XD];   // z column
            biL[tid] = bi[i * HD + tid];
            wfL[tid] = wf[i * HD + tid];
        }
        __syncthreads();

        // ---- A1 = out * M[:,i]  (diag of M already zero) ----
        v2f a1[8];
#pragma unroll
        for (int c = 0; c < 8; ++c) {
            int k = 4 * c + 2 * hh;
            a1[c].x = out_a[c].x * Mt[i * 32 + k];
            a1[c].y = out_a[c].y * Mt[i * 32 + k + 1];
        }
        float zr[8];
#pragma unroll
        for (int v = 0; v < 8; ++v)
            zr[v] = z[(size_t)(rowBase + v + 8 * hh) * XD + i];

        // ---- GEMM1: h0[16,128] = relu(xm @ WiT + z*Wiz + bi) ----
#pragma unroll
        for (int t = 0; t < 8; ++t) {
            int n = 16 * t + ln;
            float wzn = Wiz[n], bbn = biL[n];
            v8f acc;
#pragma unroll
            for (int v = 0; v < 8; ++v) acc[v] = zr[v] * wzn + bbn;
#pragma unroll
            for (int c = 0; c < 8; ++c) {
                v2f bm = Wip[(2 * c + hh) * 128 + n];
                acc = __builtin_amdgcn_wmma_f32_16x16x4_f32(
                    false, a1[c], false, bm, (short)0, acc, false, false);
            }
#pragma unroll
            for (int v = 0; v < 8; ++v)
                hscr[wave][(v + 8 * hh) * HD + n] = fmaxf(acc[v], 0.0f);
        }

        // ---- GEMM2: h1 = relu(h0 @ W1T + b1) ----
        v2f a2[32];                      // A fragments (K=128) from LDS transpose
#pragma unroll
        for (int c = 0; c < 32; ++c)
            a2[c] = *(const v2f*)&hscr[wave][ln * HD + 4 * c + 2 * hh];
#pragma unroll
        for (int t = 0; t < 8; ++t) {
            int n = 16 * t + ln;
            float bbn = b1L[n];
            v8f acc;
#pragma unroll
            for (int v = 0; v < 8; ++v) acc[v] = bbn;
#pragma unroll
            for (int c = 0; c < 32; ++c) {
                v2f bm = W1p[(2 * c + hh) * 128 + n];
                acc = __builtin_amdgcn_wmma_f32_16x16x4_f32(
                    false, a2[c], false, bm, (short)0, acc, false, false);
            }
#pragma unroll
            for (int v = 0; v < 8; ++v)
                hscr[wave][(v + 8 * hh) * HD + n] = fmaxf(acc[v], 0.0f);
        }

        // ---- GEMM3: h2 = relu(h1 @ W2T + b2); fused dot with wf_i ----
        v2f a3[32];
#pragma unroll
        for (int c = 0; c < 32; ++c)
            a3[c] = *(const v2f*)&hscr[wave][ln * HD + 4 * c + 2 * hh];
        float s[8];
#pragma unroll
        for (int v = 0; v < 8; ++v) s[v] = 0.0f;
#pragma unroll
        for (int t = 0; t < 8; ++t) {
            int n = 16 * t + ln;
            float bbn = b2L[n];
            v8f acc;
#pragma unroll
            for (int v = 0; v < 8; ++v) acc[v] = bbn;
#pragma unroll
            for (int c = 0; c < 32; ++c) {
                v2f bm = W2p[(2 * c + hh) * 128 + n];
                acc = __builtin_amdgcn_wmma_f32_16x16x4_f32(
                    false, a3[c], false, bm, (short)0, acc, false, false);
            }
            float wv = wfL[n];
#pragma unroll
            for (int v = 0; v < 8; ++v)
                s[v] += fmaxf(acc[v], 0.0f) * wv;
        }

        // ---- reduce dot over the 16 N-lanes (stays within each half) ----
#pragma unroll
        for (int off = 1; off < 16; off <<= 1) {
#pragma unroll
            for (int v = 0; v < 8; ++v)
                s[v] += __shfl_xor(s[v], off, 32);
        }
        float bfi = bf[i];
        if (ln == 0) {
#pragma unroll
            for (int v = 0; v < 8; ++v)
                valL[wave][v + 8 * hh] = 1.0f / (1.0f + __expf(-(s[v] + bfi)));
        }
        float newv = valL[wave][ln];   // per-wave LDS is in-order: safe RAW

        // ---- out[:, i] = val  (merge into A-layout registers via selects) ----
#pragma unroll
        for (int c = 0; c < 8; ++c) {
            int kx = 4 * c + 2 * hh;
            out_a[c].x = (kx     == i) ? newv : out_a[c].x;
            out_a[c].y = (kx + 1 == i) ? newv : out_a[c].y;
        }
        __syncthreads();   // protect Wip/Wiz/biL/wfL before next step restage
    }

    // ---- write final out tile ----
#pragma unroll
    for (int c = 0; c < 8; ++c)
        *(v2f*)(out + (size_t)(rowBase + ln) * XD + 4 * c + 2 * hh) = out_a[c];
}

extern "C" void kernel_launch(void* const* d_in, const int* in_sizes, int n_in,
                              void* d_out, int out_size, void* d_ws, size_t ws_size,
                              hipStream_t stream) {
    const float* x  = (const float*)d_in[0];
    const float* z  = (const float*)d_in[1];
    const float* M  = (const float*)d_in[2];
    const float* Wi = (const float*)d_in[3];
    const float* bi = (const float*)d_in[4];
    const float* wf = (const float*)d_in[5];
    const float* bf = (const float*)d_in[6];
    const float* W1 = (const float*)d_in[7];
    const float* b1 = (const float*)d_in[8];
    const float* W2 = (const float*)d_in[9];
    const float* b2 = (const float*)d_in[10];
    float* out = (float*)d_out;

    dim3 grid(BATCH / ROWS_PER_BLOCK);   // 512 blocks
    dim3 block(THREADS);                  // 8 waves
    causal_gen_kernel<<<grid, block, 0, stream>>>(x, z, M, Wi, bi, wf, bf,
                                                  W1, b1, W2, b2, out);
}